// getMaxScore_18107582120034
// MI455X (gfx1250) — compile-verified
//
#include <hip/hip_runtime.h>
#include <math.h>

typedef __attribute__((ext_vector_type(2))) float v2f;
typedef __attribute__((ext_vector_type(4))) float v4f;
typedef __attribute__((ext_vector_type(8))) float v8f;

#define D_MODEL 1024
#define HID     4096
#define N_KEYS  65536
#define TOP_K   6
#define HCHUNKS 16          // h-split for proj_v partials

// ---------------------------------------------------------------------------
// Kernel 1: q[h] = dot(Wq[h,:], query) + bq[h]      (one wave per output row)
// ---------------------------------------------------------------------------
__global__ __launch_bounds__(256) void proj_q_kernel(
    const float* __restrict__ Wq, const float* __restrict__ query,
    const float* __restrict__ bq, float* __restrict__ q) {
  __shared__ float qs[D_MODEL];
  const int tid = threadIdx.x;
  ((float4*)qs)[tid] = ((const float4*)query)[tid];   // 256 x 4 floats
  __syncthreads();

  const int lane = tid & 31;
  const int wave = tid >> 5;
  const int row  = blockIdx.x * 8 + wave;             // 512 blocks * 8 waves
  const float* __restrict__ w = Wq + (size_t)row * D_MODEL;

  float acc = 0.0f;
  #pragma unroll
  for (int i = lane * 4; i < D_MODEL; i += 32 * 4) {
    float4 wv = *(const float4*)(w + i);
    float4 xv = *(const float4*)(qs + i);
    acc += wv.x * xv.x + wv.y * xv.y + wv.z * xv.z + wv.w * xv.w;
  }
  for (int off = 16; off > 0; off >>= 1) acc += __shfl_xor(acc, off, 32);
  if (lane == 0) q[row] = acc + bq[row];
}

// ---------------------------------------------------------------------------
// Kernel 2a: partial[c][d] = sum_{h in chunk c} Wk[h][d] * q[h]
//   64 blocks (16 h-chunks x 4 d-stripes) -> 16x more memory parallelism than
//   a single column walk; deterministic (no float atomics).
// ---------------------------------------------------------------------------
__global__ __launch_bounds__(256) void proj_v_partial_kernel(
    const float* __restrict__ Wk, const float* __restrict__ q,
    float* __restrict__ partial) {
  __shared__ float qs[256];
  const int tid    = threadIdx.x;
  const int hchunk = blockIdx.x >> 2;                 // 0..15
  const int dblk   = blockIdx.x & 3;                  // 0..3
  qs[tid] = q[hchunk * 256 + tid];
  __syncthreads();

  const int d = dblk * 256 + tid;
  const float* __restrict__ w = Wk + (size_t)(hchunk * 256) * D_MODEL + d;
  float acc = 0.0f;
  #pragma unroll 8
  for (int h = 0; h < 256; ++h)
    acc += w[(size_t)h * D_MODEL] * qs[h];
  partial[hchunk * D_MODEL + d] = acc;
}

// Kernel 2b: v[d] = sum_c partial[c][d]
__global__ __launch_bounds__(256) void proj_v_reduce_kernel(
    const float* __restrict__ partial, float* __restrict__ v) {
  const int d = blockIdx.x * 256 + threadIdx.x;       // 4 blocks -> 1024
  float s = 0.0f;
  #pragma unroll
  for (int c = 0; c < HCHUNKS; ++c)
    s += partial[c * D_MODEL + d];
  v[d] = s;
}

// ---------------------------------------------------------------------------
// Kernel 3: scores[n] = key_mat[n,:] . v     via V_WMMA_F32_16X16X4_F32
//   one wave handles 16 rows; B = v chunk broadcast to all 16 columns.
//   K permuted per-WMMA so each lane's A operand is one contiguous float4
//   (lane<16 supplies K-slots {0,1}, lane>=16 supplies {2,3}).
//   key_mat (256MB > L2) loaded non-temporal.
// ---------------------------------------------------------------------------
__global__ __launch_bounds__(256) void scores_wmma_kernel(
    const float* __restrict__ key, const float* __restrict__ v,
    float* __restrict__ scores) {
  __shared__ float vs[D_MODEL];
  const int tid = threadIdx.x;
  ((float4*)vs)[tid] = ((const float4*)v)[tid];
  __syncthreads();

  const int lane    = tid & 31;
  const int wave    = tid >> 5;
  const int rowbase = (blockIdx.x * 8 + wave) * 16;   // 512 blocks * 8 * 16 rows
  const int m       = lane & 15;
  const int khalf   = (lane >> 4) * 4;                // 0 or 4

  const float* __restrict__ krow = key + (size_t)(rowbase + m) * D_MODEL + khalf;
  const float* __restrict__ vrow = vs + khalf;

  v8f c = {};
  #pragma unroll 4
  for (int kb = 0; kb < D_MODEL; kb += 16) {
    v4f a0 = __builtin_nontemporal_load((const v4f*)(krow + kb));
    v4f a1 = __builtin_nontemporal_load((const v4f*)(krow + kb + 8));
    v4f b0 = *(const v4f*)(vrow + kb);
    v4f b1 = *(const v4f*)(vrow + kb + 8);

    v2f A, B;
    A[0] = a0[0]; A[1] = a0[1]; B[0] = b0[0]; B[1] = b0[1];
    c = __builtin_amdgcn_wmma_f32_16x16x4_f32(false, A, false, B, (short)0, c, false, false);
    A[0] = a0[2]; A[1] = a0[3]; B[0] = b0[2]; B[1] = b0[3];
    c = __builtin_amdgcn_wmma_f32_16x16x4_f32(false, A, false, B, (short)0, c, false, false);
    A[0] = a1[0]; A[1] = a1[1]; B[0] = b1[0]; B[1] = b1[1];
    c = __builtin_amdgcn_wmma_f32_16x16x4_f32(false, A, false, B, (short)0, c, false, false);
    A[0] = a1[2]; A[1] = a1[3]; B[0] = b1[2]; B[1] = b1[3];
    c = __builtin_amdgcn_wmma_f32_16x16x4_f32(false, A, false, B, (short)0, c, false, false);
  }

  // D layout: lane 0 VGPR g = D[M=g][N=0]; lane 16 VGPR g = D[M=8+g][N=0].
  if (lane == 0) {
    float4 s0 = { c[0], c[1], c[2], c[3] };
    float4 s1 = { c[4], c[5], c[6], c[7] };
    *(float4*)(scores + rowbase)     = s0;
    *(float4*)(scores + rowbase + 4) = s1;
  }
  if (lane == 16) {
    float4 s0 = { c[0], c[1], c[2], c[3] };
    float4 s1 = { c[4], c[5], c[6], c[7] };
    *(float4*)(scores + rowbase + 8)  = s0;
    *(float4*)(scores + rowbase + 12) = s1;
  }
}

// ---------------------------------------------------------------------------
// Kernel 4: 6 sequential masked-argmax passes (== the torch loop / top_k)
// ---------------------------------------------------------------------------
__global__ __launch_bounds__(256) void top6_kernel(
    const float* __restrict__ scores, int* __restrict__ idx_out) {
  __shared__ float sval[256];
  __shared__ int   sidx[256];
  __shared__ int   chosen[TOP_K];
  const int tid = threadIdx.x;

  for (int p = 0; p < TOP_K; ++p) {
    float best = -INFINITY;
    int   bidx = 0x7fffffff;
    for (int i = tid; i < N_KEYS; i += 256) {
      bool skip = false;
      for (int j = 0; j < p; ++j) skip = skip || (chosen[j] == i);
      if (skip) continue;
      float sv = scores[i];
      if (sv > best || (sv == best && i < bidx)) { best = sv; bidx = i; }
    }
    sval[tid] = best; sidx[tid] = bidx;
    __syncthreads();
    for (int s = 128; s > 0; s >>= 1) {
      if (tid < s) {
        float ov = sval[tid + s]; int oi = sidx[tid + s];
        if (ov > sval[tid] || (ov == sval[tid] && oi < sidx[tid])) {
          sval[tid] = ov; sidx[tid] = oi;
        }
      }
      __syncthreads();
    }
    if (tid == 0) { chosen[p] = sidx[0]; idx_out[p] = sidx[0]; }
    __syncthreads();
  }
}

// ---------------------------------------------------------------------------
// Kernel 5: out[d] = mean over 6 selected key rows
// ---------------------------------------------------------------------------
__global__ __launch_bounds__(256) void gather_mean_kernel(
    const float* __restrict__ key, const int* __restrict__ idx,
    float* __restrict__ out) {
  const int d = blockIdx.x * 256 + threadIdx.x;       // 4 blocks -> 1024
  float s = 0.0f;
  #pragma unroll
  for (int j = 0; j < TOP_K; ++j)
    s += key[(size_t)idx[j] * D_MODEL + d];
  out[d] = s * (1.0f / 6.0f);
}

// ---------------------------------------------------------------------------
extern "C" void kernel_launch(void* const* d_in, const int* in_sizes, int n_in,
                              void* d_out, int out_size, void* d_ws, size_t ws_size,
                              hipStream_t stream) {
  const float* query   = (const float*)d_in[0];   // [1024]
  const float* key_mat = (const float*)d_in[1];   // [65536,1024]
  const float* Wq      = (const float*)d_in[2];   // [4096,1024]
  const float* bq      = (const float*)d_in[3];   // [4096]
  const float* Wk      = (const float*)d_in[4];   // [4096,1024]
  // bk (d_in[5]) adds the same constant bk.q to every score -> cannot change
  // the top-k selection; it is dropped (output depends only on selected rows).

  float* ws      = (float*)d_ws;
  float* q       = ws;                                  // 4096 floats
  float* v       = ws + HID;                            // 1024 floats
  float* scores  = ws + HID + D_MODEL;                  // 65536 floats
  float* partial = ws + HID + D_MODEL + N_KEYS;         // 16*1024 floats
  int*   idx     = (int*)(partial + HCHUNKS * D_MODEL); // 6 ints

  float* out = (float*)d_out;                           // [1024]

  proj_q_kernel        <<<HID / 8,           256, 0, stream>>>(Wq, query, bq, q);
  proj_v_partial_kernel<<<HCHUNKS * 4,       256, 0, stream>>>(Wk, q, partial);
  proj_v_reduce_kernel <<<D_MODEL / 256,     256, 0, stream>>>(partial, v);
  scores_wmma_kernel   <<<N_KEYS / (8 * 16), 256, 0, stream>>>(key_mat, v, scores);
  top6_kernel          <<<1,                 256, 0, stream>>>(scores, idx);
  gather_mean_kernel   <<<D_MODEL / 256,     256, 0, stream>>>(key_mat, idx, out);
}